// ConsistentHashEmbedder_11768210391673
// MI455X (gfx1250) — compile-verified
//
#include <hip/hip_runtime.h>
#include <cmath>

#define NL      16
#define LOG2T   19
#define T_SIZE  (1 << LOG2T)
#define HMASK   (T_SIZE - 1)
#define HPRIME  2654435761u
#define IMGF    2048.0f
#define PIV     1024

#define AS1 __attribute__((address_space(1)))
#define AS3 __attribute__((address_space(3)))

#if defined(__has_builtin)
#  if __has_builtin(__builtin_amdgcn_global_load_async_to_lds_b32)
#    define HAVE_ASYNC_LDS 1
#  else
#    define HAVE_ASYNC_LDS 0
#  endif
#  if __has_builtin(__builtin_amdgcn_s_wait_asynccnt)
#    define HAVE_WAIT_ASYNC 1
#  else
#    define HAVE_WAIT_ASYNC 0
#  endif
#else
#  define HAVE_ASYNC_LDS 0
#  define HAVE_WAIT_ASYNC 0
#endif

struct ResArgs { int res[NL]; };

// ---------------------------------------------------------------------------
// Kernel 1: build direct consistent-hash map:  hmap[li*T + h] = tv[lower_bound(tk, 2h)]
// LDS pivot table (async global->LDS gather) cuts the global binary search
// from ~22 dependent loads to ~12 (+10 LDS steps).
// ---------------------------------------------------------------------------
__global__ __launch_bounds__(256)
void cdna5_build_hmap(const int* __restrict__ tk, const int* __restrict__ tv,
                      const int* __restrict__ lengths, int maxP,
                      int* __restrict__ hmap)
{
    __shared__ int s_piv[PIV];
    const int li = blockIdx.y;
    const int n  = lengths[li];                       // uniform
    const int* tkr = tk + (size_t)li * (size_t)maxP;
    const int* tvr = tv + (size_t)li * (size_t)maxP;
    const int step = (n + PIV - 1) / PIV;             // window size
    const int m    = (n + step - 1) / step;           // #pivots (<= PIV)

    for (int j = threadIdx.x; j < m; j += blockDim.x) {
#if HAVE_ASYNC_LDS
        __builtin_amdgcn_global_load_async_to_lds_b32(
            (AS1 int*)(tkr + (size_t)j * (size_t)step),
            (AS3 int*)&s_piv[j], 0, 0);
#else
        s_piv[j] = tkr[(size_t)j * (size_t)step];
#endif
    }
#if HAVE_ASYNC_LDS
#  if HAVE_WAIT_ASYNC
    __builtin_amdgcn_s_wait_asynccnt(0);
#  else
    asm volatile("s_wait_asynccnt 0" ::: "memory");
#  endif
#endif
    __syncthreads();

    const int h = blockIdx.x * blockDim.x + threadIdx.x;   // grid covers [0,T)
    const int q = 2 * h;

    // lower_bound over pivots (LDS)
    int lo = 0, hi = m;
    while (lo < hi) {
        int mid = (lo + hi) >> 1;
        int pv  = s_piv[mid];
        lo = (pv < q) ? mid + 1 : lo;
        hi = (pv < q) ? hi      : mid;
    }
    const int jp  = lo;                                // first pivot >= q (or m)
    int glo = (jp == 0) ? 0 : (jp - 1) * step + 1;
    int ghi = (jp <  m) ? jp * step : n;

    // lower_bound over tk[glo, ghi) (global, L2-resident)
    while (glo < ghi) {
        int mid = (glo + ghi) >> 1;
        int kv  = tkr[mid];
        glo = (kv < q) ? mid + 1 : glo;
        ghi = (kv < q) ? ghi     : mid;
    }
    int k = glo;
    if (k == n) k = 0;                                 // circular wrap
    hmap[((size_t)li << LOG2T) + (size_t)h] = tvr[k];
}

// ---------------------------------------------------------------------------
// Fallback searchsorted lookup (used only if workspace too small for hmap)
// ---------------------------------------------------------------------------
__device__ __forceinline__
int cdna5_search_lookup(unsigned h, const int* __restrict__ tkr,
                        const int* __restrict__ tvr, int n)
{
    const int q = 2 * (int)h;
    int lo = 0, hi = n;
    while (lo < hi) {
        int mid = (lo + hi) >> 1;
        int kv  = tkr[mid];
        lo = (kv < q) ? mid + 1 : lo;
        hi = (kv < q) ? hi      : mid;
    }
    if (lo == n) lo = 0;
    return tvr[lo];
}

// ---------------------------------------------------------------------------
// Kernel 2: main embedding. 1 thread = 1 point. 16 levels unrolled.
// Random b64 gathers stay in the 192MB L2 (tables 64MB + hmap 8MB).
// ---------------------------------------------------------------------------
__global__ __launch_bounds__(256)
void cdna5_hash_embed(const float* __restrict__ x,
                      const float* __restrict__ tables,
                      const int* __restrict__ hmap,      // may be null
                      const int* __restrict__ tk,
                      const int* __restrict__ tv,
                      const int* __restrict__ lengths,
                      int maxP, int B,
                      float* __restrict__ outF,
                      int* __restrict__ outI,
                      ResArgs ra)
{
    const int p = blockIdx.x * blockDim.x + threadIdx.x;
    if (p >= B) return;

    const float2 xv = ((const float2*)x)[p];
    const float x0 = xv.x, x1 = xv.y;

    float feats[2 * NL];
    int li = 0;

#pragma unroll
    for (int i = 0; i < NL; ++i) {
        const int   res = ra.res[i];
        const float gs  = IMGF / (float)res;
        const float f0  = floorf(x0 / gs);
        const float f1  = floorf(x1 / gs);
        const int   b0  = (int)f0;
        const int   b1  = (int)f1;
        // replicate reference: mn = bl*gs; mx = mn+gs; w = (x-mn)/(mx-mn)
        const float mn0 = f0 * gs, mn1 = f1 * gs;
        const float w0  = (x0 - mn0) / ((mn0 + gs) - mn0);
        const float w1  = (x1 - mn1) / ((mn1 + gs) - mn1);

        int i00, i01, i10, i11;
        const bool isHash = ((res + 1) * (res + 1)) > T_SIZE;   // uniform
        if (isHash) {
            const unsigned u0 = (unsigned)b0, u1 = (unsigned)b1;
            const unsigned h00 = (u0        ^ ( u1        * HPRIME)) & HMASK;
            const unsigned h01 = (u0        ^ ((u1 + 1u)  * HPRIME)) & HMASK;
            const unsigned h10 = ((u0 + 1u) ^ ( u1        * HPRIME)) & HMASK;
            const unsigned h11 = ((u0 + 1u) ^ ((u1 + 1u)  * HPRIME)) & HMASK;
            if (hmap) {
                const int* hm = hmap + ((size_t)li << LOG2T);
                i00 = hm[h00]; i01 = hm[h01]; i10 = hm[h10]; i11 = hm[h11];
            } else {
                const int* tkr = tk + (size_t)li * (size_t)maxP;
                const int* tvr = tv + (size_t)li * (size_t)maxP;
                const int  n   = lengths[li];
                i00 = cdna5_search_lookup(h00, tkr, tvr, n);
                i01 = cdna5_search_lookup(h01, tkr, tvr, n);
                i10 = cdna5_search_lookup(h10, tkr, tvr, n);
                i11 = cdna5_search_lookup(h11, tkr, tvr, n);
            }
            ++li;
        } else {
            i00 = b0 * res + b1;
            i01 = i00 + 1;
            i10 = i00 + res;
            i11 = i10 + 1;
        }

        const float2* tb = (const float2*)(tables + (((size_t)i << LOG2T) << 1));
        const float2 e00 = tb[i00];
        const float2 e01 = tb[i01];
        const float2 e10 = tb[i10];
        const float2 e11 = tb[i11];

        const float om1 = 1.0f - w1, om0 = 1.0f - w0;
        const float c0x = e00.x * om1 + e01.x * w1;
        const float c0y = e00.y * om1 + e01.y * w1;
        const float c1x = e10.x * om1 + e11.x * w1;
        const float c1y = e10.y * om1 + e11.y * w1;
        feats[2 * i + 0] = c0x * om0 + c1x * w0;
        feats[2 * i + 1] = c0y * om0 + c1y * w0;

        *(int4*)(outI + (((size_t)i * (size_t)B + (size_t)p) << 2)) =
            make_int4(i00, i01, i10, i11);
    }

    float4* of = (float4*)(outF + (size_t)p * (size_t)(2 * NL));
#pragma unroll
    for (int j = 0; j < (2 * NL) / 4; ++j)
        of[j] = make_float4(feats[4 * j + 0], feats[4 * j + 1],
                            feats[4 * j + 2], feats[4 * j + 3]);
}

// ---------------------------------------------------------------------------
extern "C" void kernel_launch(void* const* d_in, const int* in_sizes, int n_in,
                              void* d_out, int out_size, void* d_ws, size_t ws_size,
                              hipStream_t stream)
{
    const float* x       = (const float*)d_in[0];
    const float* tables  = (const float*)d_in[1];
    const int*   tk      = (const int*)d_in[2];
    const int*   tv      = (const int*)d_in[3];
    const int*   lengths = (const int*)d_in[4];

    const int B    = in_sizes[0] / 2;
    const int L    = in_sizes[4];
    const int maxP = (L > 0) ? in_sizes[2] / L : 0;

    float* outF = (float*)d_out;
    int*   outI = (int*)d_out + (size_t)(2 * NL) * (size_t)B;

    // Replicate numpy float32 RES computation:
    //   b = float32(exp((log(2048)-log(16))/15)); res[i] = floor(float32(16 * b^i))
    ResArgs ra;
    {
        const double bd = std::exp((std::log(2048.0) - std::log(16.0)) / 15.0);
        const float  bf = (float)bd;
        for (int i = 0; i < NL; ++i) {
            const float pw = (float)std::pow((double)bf, (double)i);
            ra.res[i] = (int)std::floor(16.0f * pw);
        }
    }

    int* hmap = (int*)d_ws;
    const bool use_map =
        (L > 0) && (ws_size >= (size_t)L * (size_t)T_SIZE * sizeof(int));

    if (use_map) {
        dim3 grid(T_SIZE / 256, L);
        cdna5_build_hmap<<<grid, 256, 0, stream>>>(tk, tv, lengths, maxP, hmap);
    }

    const int nb = (B + 255) / 256;
    cdna5_hash_embed<<<nb, 256, 0, stream>>>(
        x, tables, use_map ? hmap : nullptr,
        tk, tv, lengths, maxP, B, outF, outI, ra);
}